// CrossThresholdAttention_35098472743308
// MI455X (gfx1250) — compile-verified
//
#include <hip/hip_runtime.h>

#define Tn 8
#define Bn 64
#define Ln 1024

static constexpr int    BLT = Bn * Ln * Tn;                       // 524288
static constexpr size_t BLL = (size_t)Bn * Ln * Ln;               // 67108864

typedef float v2f __attribute__((ext_vector_type(2)));
typedef float v8f __attribute__((ext_vector_type(8)));

// ---------------------------------------------------------------------------
// Kernel 1: six fused linear projections  out[p][b,l,u] = sum_t X_p[bl,t]*W_p[u,t] + bias_p[u]
// p: 0=q_g(gpcm) 1=k_c(coral) 2=v_c(coral) 3=q_c(coral) 4=k_g(gpcm) 5=v_g(gpcm)
// ---------------------------------------------------------------------------
__global__ __launch_bounds__(256)
void cta_proj_kernel(const float* __restrict__ g, const float* __restrict__ c,
                     const float* __restrict__ Wqg, const float* __restrict__ bqg,
                     const float* __restrict__ Wkc, const float* __restrict__ bkc,
                     const float* __restrict__ Wvc, const float* __restrict__ bvc,
                     const float* __restrict__ Wqc, const float* __restrict__ bqc,
                     const float* __restrict__ Wkg, const float* __restrict__ bkg,
                     const float* __restrict__ Wvg, const float* __restrict__ bvg,
                     float* __restrict__ ws)
{
    const int p   = blockIdx.y;                          // uniform per block
    const int row = blockIdx.x * blockDim.x + threadIdx.x;   // 0 .. B*L-1

    const float* Xs[6] = {g, c, c, c, g, g};
    const float* Ws[6] = {Wqg, Wkc, Wvc, Wqc, Wkg, Wvg};
    const float* Bs[6] = {bqg, bkc, bvc, bqc, bkg, bvg};

    const float* x = Xs[p] + (size_t)row * Tn;
    const float* W = Ws[p];
    const float* bi = Bs[p];

    float4 x0 = reinterpret_cast<const float4*>(x)[0];
    float4 x1 = reinterpret_cast<const float4*>(x)[1];
    float xr[8] = {x0.x, x0.y, x0.z, x0.w, x1.x, x1.y, x1.z, x1.w};

    float o[8];
#pragma unroll
    for (int u = 0; u < 8; ++u) {
        float acc = bi[u];
#pragma unroll
        for (int t = 0; t < 8; ++t) acc += xr[t] * W[u * 8 + t];
        o[u] = acc;
    }

    float* dst = ws + (size_t)p * BLT + (size_t)row * Tn;
    reinterpret_cast<float4*>(dst)[0] = make_float4(o[0], o[1], o[2], o[3]);
    reinterpret_cast<float4*>(dst)[1] = make_float4(o[4], o[5], o[6], o[7]);
}

// ---------------------------------------------------------------------------
// Kernel 2: fused attention for one (batch, 16-row strip, direction).
// grid = (4096, 2), block = 128 (4 waves).
//   phase 1: 64 score tiles (16x16) via V_WMMA_F32_16X16X4_F32, K=8 as 2 chained
//            K=4 WMMAs; scaled scores -> 64KB LDS (XOR bank swizzle, stride 1024)
//   phase 2: per-wave softmax over 4 rows (8 lanes/row), fused attn@v accumulate
//   phase 3: coalesced normalized attention-matrix write (dominant HBM traffic)
// ---------------------------------------------------------------------------
__global__ __launch_bounds__(128)
void cta_attn_kernel(const float* __restrict__ g, const float* __restrict__ cth,
                     const float* __restrict__ ws, float* __restrict__ out)
{
    __shared__ float sc[16 * 1024];                      // exactly 64 KB

    const int tid  = threadIdx.x;
    const int lane = tid & 31;
    const int wv   = tid >> 5;                           // wave 0..3
    const int strip = blockIdx.x;                        // 0..4095
    const int b    = strip >> 6;
    const int i0   = (strip & 63) << 4;                  // row-strip base
    const int dir  = blockIdx.y;                         // 0: g->c, 1: c->g

    const float* q = ws + (size_t)(dir ? 3 : 0) * BLT;
    const float* k = ws + (size_t)(dir ? 4 : 1) * BLT;
    const float* v = ws + (size_t)(dir ? 5 : 2) * BLT;
    const float* x = dir ? cth : g;
    float* attn_out = out + (size_t)2 * BLT + (size_t)dir * BLL + (size_t)b * Ln * Ln;
    float* att_out  = out + (size_t)dir * BLT + ((size_t)(b * Ln + i0)) * Tn;

    // ---- A matrix (16x8 q rows) in 16x16x4 f32 layout: two K-chunks ----
    const int arow = lane & 15;                 // A: M index / B: N index
    const int kb   = (lane >> 4) << 1;          // K pair base: 0 or 2
    const float* qr = q + ((size_t)(b * Ln + i0 + arow)) * Tn;
    v2f a0, a1;
    a0.x = qr[kb];     a0.y = qr[kb + 1];
    a1.x = qr[4 + kb]; a1.y = qr[4 + kb + 1];

    const float inv_scale = 0.35355339059327373f;   // 1/sqrt(8)
    const int n  = lane & 15;
    const int mh = (lane >> 4) << 3;

    for (int it = 0; it < 16; ++it) {
        const int j0 = (wv + (it << 2)) << 4;
        const float* kr = k + ((size_t)(b * Ln + j0 + arow)) * Tn;
        v2f b0, b1;
        b0.x = kr[kb];     b0.y = kr[kb + 1];
        b1.x = kr[4 + kb]; b1.y = kr[4 + kb + 1];
        v8f cc = {};
        cc = __builtin_amdgcn_wmma_f32_16x16x4_f32(false, a0, false, b0,
                                                   (short)0, cc, false, false);
        cc = __builtin_amdgcn_wmma_f32_16x16x4_f32(false, a1, false, b1,
                                                   (short)0, cc, false, false);
        // C/D layout: vgpr r -> (M = r + 8*(lane>=16), N = lane&15)
#pragma unroll
        for (int r = 0; r < 8; ++r) {
            const int m  = mh + r;
            const int jc = j0 + n;
            sc[(m << 10) + (jc ^ (m << 2))] = cc[r] * inv_scale;   // XOR swizzle
        }
    }
    __syncthreads();

    // ---- softmax: wave wv owns rows 4wv..4wv+3, 8 lanes per row ----
    const int rr  = lane >> 3;
    const int sub = lane & 7;
    const int m   = (wv << 2) + rr;
    const int swz = m << 2;
    float* srow = sc + (m << 10);

    float mx = -__builtin_inff();
    for (int tt = 0; tt < 128; ++tt) {
        const int j = sub + (tt << 3);
        mx = fmaxf(mx, srow[j ^ swz]);
    }
    mx = fmaxf(mx, __shfl_xor(mx, 1));
    mx = fmaxf(mx, __shfl_xor(mx, 2));
    mx = fmaxf(mx, __shfl_xor(mx, 4));

    float sum = 0.f;
    float acc[8] = {0.f, 0.f, 0.f, 0.f, 0.f, 0.f, 0.f, 0.f};
    for (int tt = 0; tt < 128; ++tt) {
        const int j = sub + (tt << 3);
        const float e = __expf(srow[j ^ swz] - mx);
        srow[j ^ swz] = e;                         // stash unnormalized prob
        sum += e;
        const float* vr = v + ((size_t)(b * Ln + j)) * Tn;
#pragma unroll
        for (int t2 = 0; t2 < 8; ++t2) acc[t2] += e * vr[t2];
    }
    sum += __shfl_xor(sum, 1);
    sum += __shfl_xor(sum, 2);
    sum += __shfl_xor(sum, 4);
    const float inv = 1.0f / sum;
#pragma unroll
    for (int t2 = 0; t2 < 8; ++t2) {
        acc[t2] += __shfl_xor(acc[t2], 1);
        acc[t2] += __shfl_xor(acc[t2], 2);
        acc[t2] += __shfl_xor(acc[t2], 4);
    }
    if (sub == 0) {
        const float* xr = x + ((size_t)(b * Ln + i0 + m)) * Tn;
        float* o = att_out + m * Tn;
#pragma unroll
        for (int t2 = 0; t2 < 8; ++t2) o[t2] = xr[t2] + acc[t2] * inv;
    }
    __syncthreads();

    // ---- coalesced normalized attention write: wave writes its 4 rows ----
    for (int itr = 0; itr < 128; ++itr) {
        const int idx = (itr << 5) + lane;         // 0..4095 over 4 rows
        const int mr  = idx >> 10;                 // row within wave (uniform/iter)
        const int j   = idx & 1023;
        const float invm = __shfl(inv, mr << 3);   // lane 8*mr holds that row's inv
        const int mm  = (wv << 2) + mr;
        const float pj = sc[(mm << 10) + (j ^ (mm << 2))] * invm;
        attn_out[(((size_t)(i0 + mm)) << 10) + j] = pj;
    }
}

// ---------------------------------------------------------------------------
extern "C" void kernel_launch(void* const* d_in, const int* in_sizes, int n_in,
                              void* d_out, int out_size, void* d_ws, size_t ws_size,
                              hipStream_t stream)
{
    const float* g   = (const float*)d_in[0];
    const float* c   = (const float*)d_in[1];
    const float* Wqg = (const float*)d_in[2];  const float* bqg = (const float*)d_in[3];
    const float* Wkc = (const float*)d_in[4];  const float* bkc = (const float*)d_in[5];
    const float* Wvc = (const float*)d_in[6];  const float* bvc = (const float*)d_in[7];
    const float* Wqc = (const float*)d_in[8];  const float* bqc = (const float*)d_in[9];
    const float* Wkg = (const float*)d_in[10]; const float* bkg = (const float*)d_in[11];
    const float* Wvg = (const float*)d_in[12]; const float* bvg = (const float*)d_in[13];
    float* ws  = (float*)d_ws;                 // needs 6*B*L*T*4 = 12.6 MB
    float* out = (float*)d_out;

    dim3 pb(256), pg((Bn * Ln) / 256, 6);
    cta_proj_kernel<<<pg, pb, 0, stream>>>(g, c, Wqg, bqg, Wkc, bkc, Wvc, bvc,
                                           Wqc, bqc, Wkg, bkg, Wvg, bvg, ws);

    dim3 ab(128), ag(4096, 2);
    cta_attn_kernel<<<ag, ab, 0, stream>>>(g, c, ws, out);
}